// TransformerQuantizerRein_76957224010050
// MI455X (gfx1250) — compile-verified
//
#include <hip/hip_runtime.h>
#include <hip/hip_bf16.h>

typedef float v2f __attribute__((ext_vector_type(2)));
typedef float v8f __attribute__((ext_vector_type(8)));

#define HWD   4096   // h*w
#define CD    256    // channels (= reduction depth of every GEMM)
#define KD    4096   // codebook entries
#define NB    4      // batch

// ---------------------------------------------------------------------------
// Row-major x Row-major^T WMMA fp32 GEMM, fully compile-time strides:
//   Out[m][n] = scale * sum_j A[m][j]*B[n][j] (+ bias[n])
//   A: [M, CD] row-major, B: [N, CD] row-major, Out row stride = ORS.
// Each wave computes a (16*BMT) x (16*BNT) tile with BMT x BNT WMMA
// accumulators (BMT=BNT=4 -> 64x64 tile, 128 acc VGPRs, 4x operand reuse).
// Fragments per ISA 7.12.2: A 16x4 -> lane(0-15)=M, half selects K pair;
// both K values contiguous in memory -> single b64 load per fragment.
// ---------------------------------------------------------------------------
template<int BMT, int BNT, long long ORS, bool HASBIAS>
__global__ __launch_bounds__(256) void wmma_gemm_rr(
    const float* __restrict__ A, const float* __restrict__ B,
    const float* __restrict__ bias, float* __restrict__ Out,
    float scale, int Mtiles, int Ntiles)
{
    const int lane = threadIdx.x & 31;
    const int tile = blockIdx.x * (blockDim.x >> 5) + (threadIdx.x >> 5);
    if (tile >= Mtiles * Ntiles) return;
    const int tm   = tile / Ntiles;
    const int tn   = tile % Ntiles;
    const int half = lane >> 4;      // 0 -> K=j..j+1, 1 -> K=j+2..j+3
    const int lm   = lane & 15;

    v8f acc[BMT][BNT];
    #pragma unroll
    for (int i = 0; i < BMT; ++i)
        #pragma unroll
        for (int j = 0; j < BNT; ++j)
            acc[i][j] = v8f{0.f,0.f,0.f,0.f,0.f,0.f,0.f,0.f};

    const float* aptr = A + (long long)(tm * 16 * BMT + lm) * CD + half * 2;
    const float* bptr = B + (long long)(tn * 16 * BNT + lm) * CD + half * 2;

    #pragma unroll 4
    for (int j = 0; j < CD; j += 4) {
        v2f a[BMT], b[BNT];
        #pragma unroll
        for (int i = 0; i < BMT; ++i)
            a[i] = *(const v2f*)(aptr + i * 16 * CD + j);   // global_load_b64
        #pragma unroll
        for (int i = 0; i < BNT; ++i)
            b[i] = *(const v2f*)(bptr + i * 16 * CD + j);
        #pragma unroll
        for (int i = 0; i < BMT; ++i)
            #pragma unroll
            for (int jj = 0; jj < BNT; ++jj)
                acc[i][jj] = __builtin_amdgcn_wmma_f32_16x16x4_f32(
                    false, a[i], false, b[jj], (short)0, acc[i][jj], false, false);
    }

    // C/D layout: VGPR r, lanes 0-15 -> M=r, lanes 16-31 -> M=r+8; col = lane&15
    #pragma unroll
    for (int jj = 0; jj < BNT; ++jj) {
        const int col = tn * 16 * BNT + jj * 16 + lm;
        const float bc = HASBIAS ? bias[col] : 0.f;
        #pragma unroll
        for (int i = 0; i < BMT; ++i) {
            #pragma unroll
            for (int r = 0; r < 8; ++r) {
                const int row = tm * 16 * BMT + i * 16 + r + half * 8;
                Out[(long long)row * ORS + col] = acc[i][jj][r] * scale + bc;
            }
        }
    }
}

// ---------------------------------------------------------------------------
// Column-major-A variant for the query projection:
//   A(m, j) = A[j*HWD + m]   (latent[n][c][s], reduction over c is strided)
//   B: [N, CD] row-major (Wq), Out row stride = CD.
// Lanes 0-15 walk contiguous m=s -> each strided load is still a coalesced
// 64B burst per half-wave.
// ---------------------------------------------------------------------------
template<int BMT, int BNT>
__global__ __launch_bounds__(256) void wmma_gemm_cm(
    const float* __restrict__ A, const float* __restrict__ B,
    const float* __restrict__ bias, float* __restrict__ Out,
    int Mtiles, int Ntiles)
{
    const int lane = threadIdx.x & 31;
    const int tile = blockIdx.x * (blockDim.x >> 5) + (threadIdx.x >> 5);
    if (tile >= Mtiles * Ntiles) return;
    const int tm   = tile / Ntiles;
    const int tn   = tile % Ntiles;
    const int half = lane >> 4;
    const int lm   = lane & 15;

    v8f acc[BMT][BNT];
    #pragma unroll
    for (int i = 0; i < BMT; ++i)
        #pragma unroll
        for (int j = 0; j < BNT; ++j)
            acc[i][j] = v8f{0.f,0.f,0.f,0.f,0.f,0.f,0.f,0.f};

    const int rowA0 = tm * 16 * BMT + lm;
    const float* bptr = B + (long long)(tn * 16 * BNT + lm) * CD + half * 2;

    #pragma unroll 2
    for (int j = 0; j < CD; j += 4) {
        const int ka = j + half * 2;
        v2f a[BMT], b[BNT];
        #pragma unroll
        for (int i = 0; i < BMT; ++i) {
            a[i].x = A[(long long)ka * HWD + rowA0 + i * 16];
            a[i].y = A[(long long)(ka + 1) * HWD + rowA0 + i * 16];
        }
        #pragma unroll
        for (int i = 0; i < BNT; ++i)
            b[i] = *(const v2f*)(bptr + i * 16 * CD + j);
        #pragma unroll
        for (int i = 0; i < BMT; ++i)
            #pragma unroll
            for (int jj = 0; jj < BNT; ++jj)
                acc[i][jj] = __builtin_amdgcn_wmma_f32_16x16x4_f32(
                    false, a[i], false, b[jj], (short)0, acc[i][jj], false, false);
    }

    #pragma unroll
    for (int jj = 0; jj < BNT; ++jj) {
        const int col = tn * 16 * BNT + jj * 16 + lm;
        const float bc = bias[col];
        #pragma unroll
        for (int i = 0; i < BMT; ++i) {
            #pragma unroll
            for (int r = 0; r < 8; ++r) {
                const int row = tm * 16 * BMT + i * 16 + r + half * 8;
                Out[(long long)row * CD + col] = acc[i][jj][r] + bc;
            }
        }
    }
}

// ---------------------------------------------------------------------------
// Online log-sum-exp over k for each (n,s); coalesced 1KB bursts per k-step.
// ---------------------------------------------------------------------------
__global__ void lse_kernel(const float* __restrict__ logit, float* __restrict__ lse)
{
    const int n = blockIdx.y;
    const int s = blockIdx.x * blockDim.x + threadIdx.x;
    const float* p = logit + (long long)n * KD * HWD + s;
    float m = -3.4e38f, sum = 0.f;
    for (int k = 0; k < KD; ++k) {
        float x = p[(long long)k * HWD];
        if (x > m) { sum = sum * __expf(m - x) + 1.f; m = x; }
        else       { sum += __expf(x - m); }
    }
    lse[n * HWD + s] = m + __logf(sum);
}

// ---------------------------------------------------------------------------
// negLogP: deterministic two-stage reduction (no float atomics).
// ---------------------------------------------------------------------------
__global__ void nlp_partial_kernel(const long long* __restrict__ code,
                                   const float* __restrict__ logit,
                                   const float* __restrict__ lse,
                                   float* __restrict__ partial)
{
    __shared__ float red[256];
    const int n = blockIdx.y;
    const int s = blockIdx.x * 256 + threadIdx.x;
    const int cd = (int)code[n * HWD + s];
    const float x = logit[((long long)n * KD + cd) * HWD + s];
    red[threadIdx.x] = lse[n * HWD + s] - x;
    __syncthreads();
    for (int off = 128; off > 0; off >>= 1) {
        if (threadIdx.x < off) red[threadIdx.x] += red[threadIdx.x + off];
        __syncthreads();
    }
    if (threadIdx.x == 0) partial[n * 16 + blockIdx.x] = red[0];
}

__global__ void nlp_final_kernel(const float* __restrict__ partial,
                                 float* __restrict__ nlp)
{
    const int n = threadIdx.x;
    if (n < NB) {
        float s = 0.f;
        for (int i = 0; i < 16; ++i) s += partial[n * 16 + i];
        nlp[n] = s;
    }
}

// ---------------------------------------------------------------------------
// quant_out[n][c][s] = value[code[n][s]][c]  (coalesced in s)
// ---------------------------------------------------------------------------
__global__ void quant_kernel(const long long* __restrict__ code,
                             const float* __restrict__ value,
                             float* __restrict__ out)
{
    const long long tid = (long long)blockIdx.x * blockDim.x + threadIdx.x;
    const int s  = (int)(tid % HWD);
    const long long nc = tid / HWD;
    const int c  = (int)(nc % CD);
    const int n  = (int)(nc / CD);
    const int cd = (int)code[n * HWD + s];
    out[tid] = value[(long long)cd * CD + c];
}

extern "C" void kernel_launch(void* const* d_in, const int* in_sizes, int n_in,
                              void* d_out, int out_size, void* d_ws, size_t ws_size,
                              hipStream_t stream) {
    (void)in_sizes; (void)n_in; (void)out_size; (void)ws_size;

    const float*      latent   = (const float*)d_in[0];      // [N,C,H,W]
    const long long*  code     = (const long long*)d_in[1];  // [N,H,W] int64
    const float*      codebook = (const float*)d_in[2];      // [K,C]
    const float*      Wq = (const float*)d_in[3];
    const float*      bq = (const float*)d_in[4];
    const float*      Wk = (const float*)d_in[5];
    const float*      bk = (const float*)d_in[6];
    const float*      Wv = (const float*)d_in[7];
    const float*      bv = (const float*)d_in[8];

    float* out      = (float*)d_out;
    float* quantOut = out;                                   // N*C*HW
    float* logitOut = out + (long long)NB * CD * HWD;        // N*K*HW (256 MB)
    float* nlpOut   = logitOut + (long long)NB * KD * HWD;   // N

    float* ws      = (float*)d_ws;
    float* Q       = ws;                                     // N*HW*C  (16 MB)
    float* keyp    = Q + (long long)NB * HWD * CD;           // K*C     (4 MB)
    float* value   = keyp + (long long)KD * CD;              // K*C     (4 MB)
    float* lse     = value + (long long)KD * CD;             // N*HW
    float* partial = lse + (long long)NB * HWD;              // N*16

    const int THREADS = 256;               // 8 waves per block
    const int WPB = THREADS / 32;
    const float invScale = 1.0f / 64.0f;   // 1/sqrt(K), K=4096

    // keyp/value = codebook @ W^T + b : M=4096, N=256, tiles 64x64
    {
        int mt = KD / 64, nt = CD / 64;
        int blocks = (mt * nt + WPB - 1) / WPB;
        wmma_gemm_rr<4, 4, CD, true><<<blocks, THREADS, 0, stream>>>(
            codebook, Wk, bk, keyp, 1.0f, mt, nt);
        wmma_gemm_rr<4, 4, CD, true><<<blocks, THREADS, 0, stream>>>(
            codebook, Wv, bv, value, 1.0f, mt, nt);
    }

    // Q_n[s][c] = sum_j latent[n][j][s]*Wq[c][j] + bq[c]
    for (int n = 0; n < NB; ++n) {
        int mt = HWD / 64, nt = CD / 64;
        int blocks = (mt * nt + WPB - 1) / WPB;
        wmma_gemm_cm<4, 4><<<blocks, THREADS, 0, stream>>>(
            latent + (long long)n * CD * HWD, Wq, bq,
            Q + (long long)n * HWD * CD, mt, nt);
    }

    // logit_out[n][k][s] = (keyp[k] . Q_n[s]) / 64 : row-major in [k][s]
    for (int n = 0; n < NB; ++n) {
        int mt = KD / 64, nt = HWD / 64;
        int blocks = (mt * nt + WPB - 1) / WPB;
        wmma_gemm_rr<4, 4, HWD, false><<<blocks, THREADS, 0, stream>>>(
            keyp, Q + (long long)n * HWD * CD, nullptr,
            logitOut + (long long)n * KD * HWD, invScale, mt, nt);
    }

    // log-sum-exp per (n, s)
    lse_kernel<<<dim3(HWD / 256, NB), 256, 0, stream>>>(logitOut, lse);

    // negLogP (deterministic two-stage)
    nlp_partial_kernel<<<dim3(HWD / 256, NB), 256, 0, stream>>>(code, logitOut, lse, partial);
    nlp_final_kernel<<<1, 32, 0, stream>>>(partial, nlpOut);

    // quantized gather
    {
        long long total = (long long)NB * CD * HWD;
        quant_kernel<<<(int)(total / 256), 256, 0, stream>>>(code, value, quantOut);
    }
}